// RTFMVDRBeamformer_81698867905091
// MI455X (gfx1250) — compile-verified
//
#include <hip/hip_runtime.h>
#include <cmath>

typedef float v2f __attribute__((ext_vector_type(2)));
typedef float v8f __attribute__((ext_vector_type(8)));

#define BB 16
#define MM 6
#define FF 257
#define TT 1500
#define NBF (BB * FF)
#define NTILE ((TT + 15) / 16)                    /* 94 (last tile partial) */
#define TCHUNK 16
#define NCHUNK ((NTILE + TCHUNK - 1) / TCHUNK)    /* 6 */
#define LASTFULL (NTILE - 1 - (NCHUNK - 1) * TCHUNK)  /* 13 full tiles in last chunk */

// ---------------------------------------------------------------------------
// Phase 1: per-(b,f) MVDR weights. One thread per (b,f).
//   - power iteration on T^2 for principal eigenvector of target SCM
//   - canonical phase fix (largest |component| made real-positive)
//   - double-precision complex Cholesky solve of N w = a  (N is Hermitian PD)
//   - bf = w / (a^H w); store 6 complex floats per (b,f) into workspace
// ---------------------------------------------------------------------------
__global__ __launch_bounds__(128) void mvdr_weights_kernel(
    const float* __restrict__ tre, const float* __restrict__ tim,
    const float* __restrict__ nre, const float* __restrict__ nim,
    float* __restrict__ wts)
{
    int idx = blockIdx.x * blockDim.x + threadIdx.x;
    if (idx >= NBF) return;
    int b = idx / FF;
    int f = idx - b * FF;
    // SCM layout (B, M, M, F): element (i,j) at base + (i*MM+j)*FF
    const size_t base = (size_t)b * MM * MM * FF + (size_t)f;

    // ---- load target SCM (fp32) -------------------------------------------
    float Tr[MM][MM], Ti[MM][MM];
#pragma unroll
    for (int i = 0; i < MM; ++i) {
#pragma unroll
        for (int j = 0; j < MM; ++j) {
            size_t o = base + (size_t)(i * MM + j) * FF;
            Tr[i][j] = tre[o];
            Ti[i][j] = tim[o];
        }
    }

    // ---- S = T * T  (squares the eigenvalue ratio -> faster convergence) ---
    float Sr[MM][MM], Si[MM][MM];
#pragma unroll
    for (int i = 0; i < MM; ++i) {
#pragma unroll
        for (int j = 0; j < MM; ++j) {
            float ar = 0.f, ai = 0.f;
#pragma unroll
            for (int k = 0; k < MM; ++k) {
                ar += Tr[i][k] * Tr[k][j] - Ti[i][k] * Ti[k][j];
                ai += Tr[i][k] * Ti[k][j] + Ti[i][k] * Tr[k][j];
            }
            Sr[i][j] = ar;
            Si[i][j] = ai;
        }
    }

    // ---- power iteration on S ---------------------------------------------
    float vr[MM], vi[MM];
#pragma unroll
    for (int m = 0; m < MM; ++m) { vr[m] = 1.0f; vi[m] = 0.0f; }
    for (int it = 0; it < 40; ++it) {
        float yr[MM], yi[MM];
        float nrm = 0.f;
#pragma unroll
        for (int i = 0; i < MM; ++i) {
            float ar = 0.f, ai = 0.f;
#pragma unroll
            for (int j = 0; j < MM; ++j) {
                ar += Sr[i][j] * vr[j] - Si[i][j] * vi[j];
                ai += Sr[i][j] * vi[j] + Si[i][j] * vr[j];
            }
            yr[i] = ar; yi[i] = ai;
            nrm += ar * ar + ai * ai;
        }
        float inv = rsqrtf(fmaxf(nrm, 1e-30f));
#pragma unroll
        for (int m = 0; m < MM; ++m) { vr[m] = yr[m] * inv; vi[m] = yi[m] * inv; }
    }

    // ---- canonical phase: rotate so largest-|.| component is real-positive -
    int   mb   = 0;
    float best = vr[0] * vr[0] + vi[0] * vi[0];
#pragma unroll
    for (int m = 1; m < MM; ++m) {
        float mag = vr[m] * vr[m] + vi[m] * vi[m];
        if (mag > best) { best = mag; mb = m; }
    }
    float imag = rsqrtf(fmaxf(best, 1e-30f));
    float pr = vr[mb] * imag, pi = -vi[mb] * imag;   // p = conj(v_mb)/|v_mb|
#pragma unroll
    for (int m = 0; m < MM; ++m) {
        float r = vr[m] * pr - vi[m] * pi;
        float i = vr[m] * pi + vi[m] * pr;
        vr[m] = r; vi[m] = i;
    }

    // ---- complex double Cholesky of noise SCM: N = L L^H ------------------
    double Lr[MM][MM], Li[MM][MM];
#pragma unroll
    for (int j = 0; j < MM; ++j) {
        double djj = (double)nre[base + (size_t)(j * MM + j) * FF];
#pragma unroll
        for (int k = 0; k < j; ++k)
            djj -= Lr[j][k] * Lr[j][k] + Li[j][k] * Li[j][k];
        double ljj = sqrt(fmax(djj, 1e-30));
        Lr[j][j] = ljj; Li[j][j] = 0.0;
        double invl = 1.0 / ljj;
#pragma unroll
        for (int i = 0; i < MM; ++i) {
            if (i > j) {
                double sr = (double)nre[base + (size_t)(i * MM + j) * FF];
                double si = (double)nim[base + (size_t)(i * MM + j) * FF];
#pragma unroll
                for (int k = 0; k < j; ++k) {
                    // L[i][k] * conj(L[j][k])
                    sr -= Lr[i][k] * Lr[j][k] + Li[i][k] * Li[j][k];
                    si -= Li[i][k] * Lr[j][k] - Lr[i][k] * Li[j][k];
                }
                Lr[i][j] = sr * invl;
                Li[i][j] = si * invl;
            }
        }
    }

    // ---- forward solve L y = a --------------------------------------------
    double yr[MM], yi[MM];
#pragma unroll
    for (int i = 0; i < MM; ++i) {
        double sr = (double)vr[i], si = (double)vi[i];
#pragma unroll
        for (int k = 0; k < i; ++k) {
            sr -= Lr[i][k] * yr[k] - Li[i][k] * yi[k];
            si -= Lr[i][k] * yi[k] + Li[i][k] * yr[k];
        }
        double inv = 1.0 / Lr[i][i];
        yr[i] = sr * inv; yi[i] = si * inv;
    }
    // ---- backward solve L^H w = y -----------------------------------------
    double wr[MM], wi[MM];
#pragma unroll
    for (int ii = MM - 1; ii >= 0; --ii) {
        double sr = yr[ii], si = yi[ii];
#pragma unroll
        for (int k = ii + 1; k < MM; ++k) {
            // conj(L[k][ii]) * w[k]
            sr -= Lr[k][ii] * wr[k] + Li[k][ii] * wi[k];
            si -= Lr[k][ii] * wi[k] - Li[k][ii] * wr[k];
        }
        double inv = 1.0 / Lr[ii][ii];
        wr[ii] = sr * inv; wi[ii] = si * inv;
    }

    // ---- denom = a^H w ; bf = w / denom -----------------------------------
    double dre = 0.0, dim = 0.0;
#pragma unroll
    for (int m = 0; m < MM; ++m) {
        dre += (double)vr[m] * wr[m] + (double)vi[m] * wi[m];
        dim += (double)vr[m] * wi[m] - (double)vi[m] * wr[m];
    }
    double dinv = 1.0 / (dre * dre + dim * dim);
    float* wp = wts + (size_t)idx * 12;
#pragma unroll
    for (int m = 0; m < MM; ++m) {
        double br = (wr[m] * dre + wi[m] * dim) * dinv;
        double bi = (wi[m] * dre - wr[m] * dim) * dinv;
        wp[2 * m + 0] = (float)br;
        wp[2 * m + 1] = (float)bi;
    }
}

// ---------------------------------------------------------------------------
// Phase 2: out[b,f,t] = sum_m conj(bf[m]) * mix[b,m,f,t], via fp32 WMMA.
// Per 16-time tile: 3 chained V_WMMA_F32_16X16X4_F32 (K = 12 interleaved
// re/im channel components).  A row 0 = Re weights, A row 8 = Im weights, so
// D row 0 (VGPR0, lanes 0-15) = Re(out) and D row 8 (VGPR0, lanes 16-31) =
// Im(out): all 32 lanes store acc[0] to a contiguous 128B interleaved-complex
// region -- branch-free, fully coalesced.  Loads are unconditional with
// immediate offsets; only the single partial tail tile clamps addresses.
// ---------------------------------------------------------------------------
__global__ __launch_bounds__(256) void mvdr_apply_kernel(
    const float* __restrict__ mix_re, const float* __restrict__ mix_im,
    const float* __restrict__ wts, float* __restrict__ out)
{
    const int lane = threadIdx.x & 31;
    const int wave = blockIdx.x * (blockDim.x >> 5) + (threadIdx.x >> 5);
    int bf    = wave / NCHUNK;
    int chunk = wave - bf * NCHUNK;
    if (bf >= NBF) return;                    // whole-wave uniform exit
    int b = bf / FF;
    int f = bf - b * FF;

    const int row   = lane & 15;              // A: M index / B: N index
    const int hi    = lane >> 4;              // 0: lanes 0-15, 1: lanes 16-31
    const int khalf = hi << 1;                // B/A K offset per lane half

    const int t0 = chunk * TCHUNK * 16;       // first time sample of chunk

    // ---- A fragments (weights) + per-lane B row base pointers -------------
    const float* w = wts + (size_t)bf * 12;   // [m][re,im]
    v2f afrag[3];
    const float* p[3][2];
#pragma unroll
    for (int s = 0; s < 3; ++s) {
#pragma unroll
        for (int v = 0; v < 2; ++v) {
            int k = 4 * s + khalf + v;
            int m = k >> 1;
            int c = k & 1;                    // 0 -> re component, 1 -> im
            float wrv = w[2 * m + 0];
            float wiv = w[2 * m + 1];
            float aval = 0.f;
            if (row == 0)      aval = c ? wiv : wrv;    // D row 0  -> Re(out)
            else if (row == 8) aval = c ? wrv : -wiv;   // D row 8  -> Im(out)
            afrag[s][v] = aval;
            const float* src = c ? mix_im : mix_re;
            p[s][v] = src + ((size_t)(b * MM + m) * FF + f) * TT + t0 + row;
        }
    }

    // per-lane output pointer: lane n -> Re slot, lane 16+n -> Im slot
    float* op = out + (size_t)bf * TT * 2 + (size_t)t0 * 2 + 2 * row + hi;

    if (chunk != NCHUNK - 1) {
        // ---- 16 full tiles, unconditional loads + stores ------------------
#pragma unroll 4
        for (int it = 0; it < TCHUNK; ++it) {
            v8f acc = {0.f, 0.f, 0.f, 0.f, 0.f, 0.f, 0.f, 0.f};
#pragma unroll
            for (int s = 0; s < 3; ++s) {
                v2f bfrag;
                bfrag[0] = p[s][0][it * 16];
                bfrag[1] = p[s][1][it * 16];
                acc = __builtin_amdgcn_wmma_f32_16x16x4_f32(
                    false, afrag[s], false, bfrag, (short)0, acc, false, false);
            }
            op[it * 32] = acc[0];
        }
    } else {
        // ---- last chunk: 13 full tiles ------------------------------------
#pragma unroll 4
        for (int it = 0; it < LASTFULL; ++it) {
            v8f acc = {0.f, 0.f, 0.f, 0.f, 0.f, 0.f, 0.f, 0.f};
#pragma unroll
            for (int s = 0; s < 3; ++s) {
                v2f bfrag;
                bfrag[0] = p[s][0][it * 16];
                bfrag[1] = p[s][1][it * 16];
                acc = __builtin_amdgcn_wmma_f32_16x16x4_f32(
                    false, afrag[s], false, bfrag, (short)0, acc, false, false);
            }
            op[it * 32] = acc[0];
        }
        // ---- partial tail tile (t = 1488..1503): clamp loads, mask store --
        {
            const int toff = LASTFULL * 16;          // 208
            int t = t0 + toff + row;                 // 1488 + row
            // clamped offset relative to p (OOB lanes re-read t = TT-1; the
            // garbage lands in D columns >= 12 which are never stored)
            int coff = (t < TT) ? toff : (TT - 1 - t0 - row);
            v8f acc = {0.f, 0.f, 0.f, 0.f, 0.f, 0.f, 0.f, 0.f};
#pragma unroll
            for (int s = 0; s < 3; ++s) {
                v2f bfrag;
                bfrag[0] = p[s][0][coff];
                bfrag[1] = p[s][1][coff];
                acc = __builtin_amdgcn_wmma_f32_16x16x4_f32(
                    false, afrag[s], false, bfrag, (short)0, acc, false, false);
            }
            if (t < TT) op[toff * 2] = acc[0];
        }
    }
}

// ---------------------------------------------------------------------------
extern "C" void kernel_launch(void* const* d_in, const int* in_sizes, int n_in,
                              void* d_out, int out_size, void* d_ws, size_t ws_size,
                              hipStream_t stream) {
    (void)in_sizes; (void)n_in; (void)out_size; (void)ws_size;
    const float* mix_re = (const float*)d_in[0];
    const float* mix_im = (const float*)d_in[1];
    const float* tre    = (const float*)d_in[2];
    const float* tim    = (const float*)d_in[3];
    const float* nre    = (const float*)d_in[4];
    const float* nim    = (const float*)d_in[5];
    float* out = (float*)d_out;
    float* wts = (float*)d_ws;                // needs NBF*12 floats (~197 KB)

    int thr1 = 128;
    int blk1 = (NBF + thr1 - 1) / thr1;
    mvdr_weights_kernel<<<blk1, thr1, 0, stream>>>(tre, tim, nre, nim, wts);

    int totalWaves    = NBF * NCHUNK;         // 24672
    int wavesPerBlock = 8;                    // 256 threads, wave32
    int blk2 = (totalWaves + wavesPerBlock - 1) / wavesPerBlock;
    mvdr_apply_kernel<<<blk2, wavesPerBlock * 32, 0, stream>>>(mix_re, mix_im, wts, out);
}